// EdgeEmbAttentionAggregator_75419625718401
// MI455X (gfx1250) — compile-verified
//
#include <hip/hip_runtime.h>
#include <hip/hip_bf16.h>

// CDNA5 / gfx1250: wave32, WMMA (not MFMA). We use V_WMMA_F32_16X16X4_F32 for
// exact-fp32 matrix math: D(16x16) = A(16x4) x B(4x16) + C.
typedef __attribute__((ext_vector_type(2))) float v2f;
typedef __attribute__((ext_vector_type(8))) float v8f;

#define KDIM      128
#define NDIM      64
#define EDIM      32
#define SNEIGH    16
#define OUTSTRIDE 160          // [x(64) | h_prime(64) | h_edge(32)]
#define BSTRIDE   132          // padded LDS row stride (floats): conflict-free b64 reads
#define ALPHA_LK  0.8f

__device__ __forceinline__ float xor_add16(float v) {
  // reduce across the 16 lanes of each half-wave (xor masks < 16 stay in-half)
  v += __shfl_xor(v, 1, 32);
  v += __shfl_xor(v, 2, 32);
  v += __shfl_xor(v, 4, 32);
  v += __shfl_xor(v, 8, 32);
  return v;
}

// Stage a 128x64 row-major weight matrix into LDS *transposed*: bst[d][k],
// row stride BSTRIDE floats (pad => the b64 fragment reads below are
// conflict-free across all 64 banks).
__device__ __forceinline__ void stage_bT(const float* __restrict__ B, float* bst,
                                         int tid, int nth) {
  for (int i = tid; i < KDIM * NDIM; i += nth) {
    int k = i >> 6;       // 0..127
    int d = i & 63;       // 0..63
    bst[d * BSTRIDE + k] = B[i];
  }
}

// One wave computes a 16x64 = (16x128)x(128x64) tile via 4 N-tiles x 32 K-steps
// of V_WMMA_F32_16X16X4_F32. A fragments come straight from global (lane lo
// reads row lo; halves take complementary K pairs, so 64B lines are fully
// consumed across consecutive K-steps). B fragments come from the transposed
// LDS image of the weight matrix.
__device__ __forceinline__ void gemm16x64(const float* __restrict__ arow,
                                          const float* __restrict__ bst,
                                          int lo, int hi,
                                          v8f& c0, v8f& c1, v8f& c2, v8f& c3) {
#pragma unroll
  for (int k = 0; k < KDIM / 4; ++k) {
    const int col = 4 * k + 2 * hi;   // A lane layout: hi half takes K+2,K+3
    v2f av = *(const v2f*)(arow + col);
    v2f b0 = *(const v2f*)(bst + (lo     ) * BSTRIDE + col);
    v2f b1 = *(const v2f*)(bst + (lo + 16) * BSTRIDE + col);
    v2f b2 = *(const v2f*)(bst + (lo + 32) * BSTRIDE + col);
    v2f b3 = *(const v2f*)(bst + (lo + 48) * BSTRIDE + col);
    c0 = __builtin_amdgcn_wmma_f32_16x16x4_f32(false, av, false, b0, (short)0, c0, false, false);
    c1 = __builtin_amdgcn_wmma_f32_16x16x4_f32(false, av, false, b1, (short)0, c1, false, false);
    c2 = __builtin_amdgcn_wmma_f32_16x16x4_f32(false, av, false, b2, (short)0, c2, false, false);
    c3 = __builtin_amdgcn_wmma_f32_16x16x4_f32(false, av, false, b3, (short)0, c3, false, false);
  }
}

// Kernel 1: X = input @ W  ->  out[:, 0:64]. One wave per 16-node tile.
__global__ void __launch_bounds__(256)
xgemm_kernel(const float* __restrict__ input, const float* __restrict__ W,
             float* __restrict__ out, int N, int ntiles) {
  extern __shared__ float bst[];
  stage_bT(W, bst, threadIdx.x, blockDim.x);
  __syncthreads();

  const int wave = threadIdx.x >> 5;
  const int lane = threadIdx.x & 31;
  const int tile = blockIdx.x * 8 + wave;
  if (tile >= ntiles) return;
  const int lo = lane & 15, hi = lane >> 4;

  const int base = tile * 16;
  int row = base + lo; if (row >= N) row = N - 1;   // clamp (N % 16 == 0 normally)
  const float* arow = input + (size_t)row * KDIM;

  v8f c0 = {}, c1 = {}, c2 = {}, c3 = {};
  gemm16x64(arow, bst, lo, hi, c0, c1, c2, c3);

  // C layout: VGPR r, lane lo -> (M = r + 8*hi, N = lo). Scatter x rows.
#pragma unroll
  for (int r = 0; r < 8; ++r) {
    const int m = base + r + 8 * hi;
    if (m < N) {
      float* o = out + (size_t)m * OUTSTRIDE + lo;
      o[0]  = c0[r];
      o[16] = c1[r];
      o[32] = c2[r];
      o[48] = c3[r];
    }
  }
}

// Kernel 2: per-node attention. One wave per node.
__global__ void __launch_bounds__(256)
attn_kernel(const float* __restrict__ neigh, const float* __restrict__ eein,
            const float* __restrict__ W2, const float* __restrict__ avec,
            float* __restrict__ out, int N) {
  extern __shared__ float bst[];
  stage_bT(W2, bst, threadIdx.x, blockDim.x);
  __syncthreads();

  const int wave = threadIdx.x >> 5;
  const int lane = threadIdx.x & 31;
  const int n = blockIdx.x * 8 + wave;
  if (n >= N) return;
  const int lo = lane & 15, hi = lane >> 4;

  // neighs(16x64) = neigh_tile(16x128) @ W2
  const float* arow = neigh + ((size_t)n * SNEIGH + lo) * KDIM;
  v8f c0 = {}, c1 = {}, c2 = {}, c3 = {};
  gemm16x64(arow, bst, lo, hi, c0, c1, c2, c3);

  // ee tile in C-compatible layout: e{t}[r] = ee[s = r + 8*hi, e = 16t + lo]
  const float* eerow = eein + (size_t)n * (SNEIGH * EDIM) + (8 * hi) * EDIM + lo;
  float e0[8], e1[8];
#pragma unroll
  for (int r = 0; r < 8; ++r) {
    e0[r] = eerow[r * EDIM];
    e1[r] = eerow[r * EDIM + 16];
  }

  // attention vector slices (per-lane columns)
  const float an0 = avec[64 + lo],  an1 = avec[80 + lo];
  const float an2 = avec[96 + lo],  an3 = avec[112 + lo];
  const float ae0 = avec[128 + lo], ae1 = avec[144 + lo];

  // xdot = x[n] . a_x, reading x from out[:,0:64] (written by kernel 1;
  // same-stream kernel ordering guarantees visibility).
  const float* xr = out + (size_t)n * OUTSTRIDE;
  float xd = xr[lane] * avec[lane] + xr[32 + lane] * avec[32 + lane];
  xd += __shfl_xor(xd, 16, 32);
  xd = xor_add16(xd);   // full 32-lane reduction (order-free xor masks)

  // scores + leaky relu; er[r] is the score for s = r + 8*hi (lane-uniform per half)
  float er[8];
#pragma unroll
  for (int r = 0; r < 8; ++r) {
    float v = c0[r] * an0 + c1[r] * an1 + c2[r] * an2 + c3[r] * an3
            + e0[r] * ae0 + e1[r] * ae1;
    v = xor_add16(v);
    float s = xd + v;
    er[r] = s > 0.0f ? s : ALPHA_LK * s;
  }

  // softmax over the 16 neighbors (8 per half-wave, one xor-16 exchange)
  float m = er[0];
#pragma unroll
  for (int r = 1; r < 8; ++r) m = fmaxf(m, er[r]);
  m = fmaxf(m, __shfl_xor(m, 16, 32));

  float p[8], sum = 0.0f;
#pragma unroll
  for (int r = 0; r < 8; ++r) { p[r] = __expf(er[r] - m); sum += p[r]; }
  sum += __shfl_xor(sum, 16, 32);
  const float inv = 1.0f / sum;
#pragma unroll
  for (int r = 0; r < 8; ++r) p[r] *= inv;

  // h_prime = attn @ neighs ; h_edge = attn @ ee (halves hold s-subsets, xor-16 folds)
  float hp0 = 0.f, hp1 = 0.f, hp2 = 0.f, hp3 = 0.f, he0 = 0.f, he1 = 0.f;
#pragma unroll
  for (int r = 0; r < 8; ++r) {
    hp0 += p[r] * c0[r];
    hp1 += p[r] * c1[r];
    hp2 += p[r] * c2[r];
    hp3 += p[r] * c3[r];
    he0 += p[r] * e0[r];
    he1 += p[r] * e1[r];
  }
  hp0 += __shfl_xor(hp0, 16, 32);
  hp1 += __shfl_xor(hp1, 16, 32);
  hp2 += __shfl_xor(hp2, 16, 32);
  hp3 += __shfl_xor(hp3, 16, 32);
  he0 += __shfl_xor(he0, 16, 32);
  he1 += __shfl_xor(he1, 16, 32);

  float* orow = out + (size_t)n * OUTSTRIDE;
  if (hi == 0) {
    orow[64 + lo]  = hp0;
    orow[80 + lo]  = hp1;
    orow[128 + lo] = he0;
  } else {
    orow[96 + lo]  = hp2;
    orow[112 + lo] = hp3;
    orow[144 + lo] = he1;
  }
}

extern "C" void kernel_launch(void* const* d_in, const int* in_sizes, int n_in,
                              void* d_out, int out_size, void* d_ws, size_t ws_size,
                              hipStream_t stream) {
  (void)n_in; (void)out_size; (void)d_ws; (void)ws_size;
  const float* input = (const float*)d_in[0];
  const float* neigh = (const float*)d_in[1];
  const float* eein  = (const float*)d_in[2];
  const float* W     = (const float*)d_in[3];
  const float* W2    = (const float*)d_in[4];
  const float* avec  = (const float*)d_in[5];
  float* out = (float*)d_out;

  const int N = in_sizes[0] / KDIM;           // 50000
  const int ntiles = (N + 15) / 16;           // 3125
  const size_t lds_bytes = (size_t)NDIM * BSTRIDE * sizeof(float);  // 33792 B

  const int wpb = 8;                          // waves per block (256 threads)
  dim3 blk(256);
  dim3 g1((ntiles + wpb - 1) / wpb);
  dim3 g2((N + wpb - 1) / wpb);

  hipLaunchKernelGGL(xgemm_kernel, g1, blk, lds_bytes, stream, input, W, out, N, ntiles);
  hipLaunchKernelGGL(attn_kernel,  g2, blk, lds_bytes, stream, neigh, eein, W2, avec, out, N);
}